// SAGE_7851200217530
// MI455X (gfx1250) — compile-verified
//
#include <hip/hip_runtime.h>

#define DHID 128
#define TM   16
#define SSTR 132   // padded LDS row stride (floats) to avoid bank conflicts

typedef __attribute__((ext_vector_type(2))) float v2f;
typedef __attribute__((ext_vector_type(8))) float v8f;

// ---------------- degree kernels ----------------
__global__ void deg_kernel(const int* __restrict__ dst, float* __restrict__ deg, int E) {
    int i = blockIdx.x * blockDim.x + threadIdx.x;
    if (i < E) atomicAdd(&deg[dst[i]], 1.0f);
}

__global__ void invdeg_kernel(const float* __restrict__ deg, float* __restrict__ invd, int n) {
    int i = blockIdx.x * blockDim.x + threadIdx.x;
    if (i < n) {
        float d = deg[i];
        invd[i] = d > 0.0f ? 1.0f / d : 1.0f;
    }
}

// ---------------- edge-parallel mean-aggregation (sum part) ----------------
// one thread per (edge, 4-float chunk); h-table is L2-resident (51MB < 192MB L2)
__global__ void aggregate_kernel(const float* __restrict__ h, const int* __restrict__ src,
                                 const int* __restrict__ dst, float* __restrict__ agg,
                                 unsigned total, int d, int chunk_shift) {
    unsigned gid = blockIdx.x * blockDim.x + threadIdx.x;
    if (gid >= total) return;
    unsigned e = gid >> chunk_shift;
    unsigned c = (gid & ((1u << chunk_shift) - 1u)) << 2;
    int s = src[e], t = dst[e];
    const float4 v = *(const float4*)(h + (size_t)s * d + c);
    float* p = agg + (size_t)t * d + c;
    atomicAdd(p + 0, v.x);
    atomicAdd(p + 1, v.y);
    atomicAdd(p + 2, v.z);
    atomicAdd(p + 3, v.w);
}

// ---------------- fused SAGE layer GEMM: out = LN(ReLU(h*Wself + (agg/deg)*Wneigh + b)) ----
// 256 threads = 8 waves; block owns 16 rows; wave w owns output cols [16w, 16w+16)
__global__ __launch_bounds__(256) void sage_gemm_kernel(
    const float* __restrict__ hin, const float* __restrict__ agg,
    const float* __restrict__ invd,
    const float* __restrict__ wself, const float* __restrict__ wneigh,
    const float* __restrict__ bias,
    const float* __restrict__ gamma, const float* __restrict__ beta,
    float* __restrict__ out, int n, int din, int do_norm)
{
    __shared__ float s_a[TM * SSTR];   // self features tile  [16 x din]
    __shared__ float s_n[TM * SSTR];   // mean-neighbor tile  [16 x din]
    __shared__ float s_o[TM * SSTR];   // output tile         [16 x 128]
    __shared__ float s_mu[TM], s_rs[TM];

    const int tid  = threadIdx.x;
    const int row0 = blockIdx.x * TM;

    // stage A tiles (self + invdeg-scaled aggregate) into LDS
    for (int i = tid; i < TM * din; i += 256) {
        int r = i / din, c = i - r * din;
        float av = 0.0f, nv = 0.0f;
        if (row0 + r < n) {
            size_t g = (size_t)(row0 + r) * din + c;
            av = hin[g];
            nv = agg[g] * invd[row0 + r];
        }
        s_a[r * SSTR + c] = av;
        s_n[r * SSTR + c] = nv;
    }
    __syncthreads();

    const int wave  = tid >> 5;
    const int lane  = tid & 31;
    const int rc    = lane & 15;           // A-row / B-col / D-col within tile
    const int kgrp  = (lane >> 4) << 1;    // 0 or 2  (K sub-offset, ISA 16x4 A layout)
    const int colg  = (wave << 4) + rc;    // global output column 0..127
    const int mbase = (lane >> 4) << 3;    // 0 or 8  (D M-half, ISA 16x16 C layout)

    v8f acc = {0.f, 0.f, 0.f, 0.f, 0.f, 0.f, 0.f, 0.f};

    for (int k = 0; k < din; k += 4) {
        const int kb = k + kgrp;
        v2f a, b;
        a[0] = s_a[rc * SSTR + kb];
        a[1] = s_a[rc * SSTR + kb + 1];
        b[0] = wself[(size_t)kb * DHID + colg];
        b[1] = wself[(size_t)(kb + 1) * DHID + colg];
        acc = __builtin_amdgcn_wmma_f32_16x16x4_f32(false, a, false, b,
                                                    (short)0, acc, false, false);
        v2f an, bn;
        an[0] = s_n[rc * SSTR + kb];
        an[1] = s_n[rc * SSTR + kb + 1];
        bn[0] = wneigh[(size_t)kb * DHID + colg];
        bn[1] = wneigh[(size_t)(kb + 1) * DHID + colg];
        acc = __builtin_amdgcn_wmma_f32_16x16x4_f32(false, an, false, bn,
                                                    (short)0, acc, false, false);
    }

    // bias + ReLU, scatter D fragments into LDS output tile
    const float bc = bias[colg];
#pragma unroll
    for (int v = 0; v < 8; ++v) {
        float val = acc[v] + bc;
        s_o[(mbase + v) * SSTR + colg] = fmaxf(val, 0.0f);
    }
    __syncthreads();

    if (do_norm) {
        if (tid < TM) {
            float s = 0.f, s2 = 0.f;
            for (int c = 0; c < DHID; ++c) {
                float v = s_o[tid * SSTR + c];
                s += v; s2 += v * v;
            }
            float mu  = s * (1.0f / DHID);
            float var = s2 * (1.0f / DHID) - mu * mu;
            s_mu[tid] = mu;
            s_rs[tid] = rsqrtf(var + 1e-5f);
        }
        __syncthreads();
        for (int i = tid; i < TM * DHID; i += 256) {
            int r = i >> 7, c = i & (DHID - 1);
            if (row0 + r < n) {
                float v = s_o[r * SSTR + c];
                v = (v - s_mu[r]) * s_rs[r] * gamma[c] + beta[c];
                out[(size_t)(row0 + r) * DHID + c] = v;
            }
        }
    } else {
        for (int i = tid; i < TM * DHID; i += 256) {
            int r = i >> 7, c = i & (DHID - 1);
            if (row0 + r < n)
                out[(size_t)(row0 + r) * DHID + c] = s_o[r * SSTR + c];
        }
    }
}

// ---------------- launch ----------------
extern "C" void kernel_launch(void* const* d_in, const int* in_sizes, int n_in,
                              void* d_out, int out_size, void* d_ws, size_t ws_size,
                              hipStream_t stream) {
    const float* x   = (const float*)d_in[0];
    const int*   src = (const int*)d_in[1];
    const int*   dst = (const int*)d_in[2];
    const float* ws0 = (const float*)d_in[3];
    const float* wn0 = (const float*)d_in[4];
    const float* b0  = (const float*)d_in[5];
    const float* ws1 = (const float*)d_in[6];
    const float* wn1 = (const float*)d_in[7];
    const float* b1  = (const float*)d_in[8];
    const float* ws2 = (const float*)d_in[9];
    const float* wn2 = (const float*)d_in[10];
    const float* b2  = (const float*)d_in[11];
    const float* g0  = (const float*)d_in[12];
    const float* be0 = (const float*)d_in[13];
    const float* g1  = (const float*)d_in[14];
    const float* be1 = (const float*)d_in[15];

    const int DIN = 64;
    const int N = in_sizes[0] / DIN;   // 100000
    const int E = in_sizes[1];         // 1000000

    char* ws = (char*)d_ws;
    size_t off = 0;
    float* agg  = (float*)(ws + off); off += (size_t)N * DHID * 4;
    float* h1   = (float*)(ws + off); off += (size_t)N * DHID * 4;
    float* h2   = (float*)(ws + off); off += (size_t)N * DHID * 4;
    float* deg  = (float*)(ws + off); off += (size_t)N * 4;
    float* invd = (float*)(ws + off); off += (size_t)N * 4;
    (void)ws_size; (void)n_in; (void)out_size;

    // degrees (shared by all 3 layers)
    hipMemsetAsync(deg, 0, (size_t)N * 4, stream);
    deg_kernel<<<(E + 255) / 256, 256, 0, stream>>>(dst, deg, E);
    invdeg_kernel<<<(N + 255) / 256, 256, 0, stream>>>(deg, invd, N);

    const int nblk = (N + TM - 1) / TM;   // 6250

    // ---- layer 0: din = 64 ----
    hipMemsetAsync(agg, 0, (size_t)N * 64 * 4, stream);
    {
        unsigned total = (unsigned)E << 4;   // E * (64/4)
        aggregate_kernel<<<(total + 255) / 256, 256, 0, stream>>>(x, src, dst, agg, total, 64, 4);
    }
    sage_gemm_kernel<<<nblk, 256, 0, stream>>>(x, agg, invd, ws0, wn0, b0, g0, be0, h1, N, 64, 1);

    // ---- layer 1: din = 128 ----
    hipMemsetAsync(agg, 0, (size_t)N * DHID * 4, stream);
    {
        unsigned total = (unsigned)E << 5;   // E * (128/4)
        aggregate_kernel<<<(total + 255) / 256, 256, 0, stream>>>(h1, src, dst, agg, total, DHID, 5);
    }
    sage_gemm_kernel<<<nblk, 256, 0, stream>>>(h1, agg, invd, ws1, wn1, b1, g1, be1, h2, N, DHID, 1);

    // ---- layer 2: din = 128, no norm ----
    hipMemsetAsync(agg, 0, (size_t)N * DHID * 4, stream);
    {
        unsigned total = (unsigned)E << 5;
        aggregate_kernel<<<(total + 255) / 256, 256, 0, stream>>>(h2, src, dst, agg, total, DHID, 5);
    }
    sage_gemm_kernel<<<nblk, 256, 0, stream>>>(h2, agg, invd, ws2, wn2, b2, nullptr, nullptr,
                                               (float*)d_out, N, DHID, 0);
}